// FeatureMetricConsistencyLoss_274877907268
// MI455X (gfx1250) — compile-verified
//
#include <hip/hip_runtime.h>
#include <hip/hip_bf16.h>
#include <math.h>

#define N_BOXES 256
#define N_VIEWS 6
#define C_FEAT  256
#define HF      116
#define WF      200
#define P_TOT   (N_BOXES * N_VIEWS)   // 1536
#define EPSV    1e-8f
#define CHW     ((size_t)C_FEAT * HF * WF)
#define HW      ((size_t)HF * WF)

typedef __attribute__((ext_vector_type(2))) float v2f;
typedef __attribute__((ext_vector_type(8))) float v8f;

struct Ws {
    int   first;                 // atomicMin target
    float stored_norm2;
    int   u0c[P_TOT];
    int   v0c[P_TOT];
    int   valid[P_TOT];
    float stored[C_FEAT];
    float norm2[P_TOT];
    float dots[P_TOT];
};

// ---------------- Kernel 1: init ----------------
__global__ void k_init(Ws* ws) {
    if (threadIdx.x == 0) ws->first = 0x7FFFFFFF;
}

// ---------------- Kernel 2: project 1536 points ----------------
__global__ void k_project(const float* __restrict__ boxes,
                          const float* __restrict__ Km,
                          const float* __restrict__ Tm,
                          const int*   __restrict__ img,
                          Ws* __restrict__ ws) {
    int p = blockIdx.x * blockDim.x + threadIdx.x;
    if (p >= P_TOT) return;
    int n = p / N_VIEWS;
    int v = p % N_VIEWS;

    float cx = boxes[n * 7 + 0];
    float cy = boxes[n * 7 + 1];
    float cz = boxes[n * 7 + 2];

    const float* Tv = Tm + v * 16;
    float pc[3];
#pragma unroll
    for (int i = 0; i < 3; ++i)
        pc[i] = Tv[i * 4 + 0] * cx + Tv[i * 4 + 1] * cy + Tv[i * 4 + 2] * cz + Tv[i * 4 + 3];

    const float* Kv = Km + v * 9;
    float p2[3];
#pragma unroll
    for (int i = 0; i < 3; ++i)
        p2[i] = Kv[i * 3 + 0] * pc[0] + Kv[i * 3 + 1] * pc[1] + Kv[i * 3 + 2] * pc[2];

    float z  = pc[2];
    float u  = p2[0] / p2[2];
    float vv = p2[1] / z;

    float Wimg = (float)img[v * 2 + 1];
    float Himg = (float)img[v * 2 + 0];
    float us = u  * (float)WF / Wimg;
    float vs = vv * (float)HF / Himg;
    float u0f = floorf(us);
    float v0f = floorf(vs);

    bool valid = (z > 0.f) && (u >= 0.f) && (u < Wimg) && (vv >= 0.f) && (vv < Himg)
                 && (u0f < (float)(WF - 1)) && (v0f < (float)(HF - 1));

    int u0c = (int)fminf(fmaxf(u0f, 0.f), (float)(WF - 2));
    int v0c = (int)fminf(fmaxf(v0f, 0.f), (float)(HF - 2));

    ws->u0c[p]   = u0c;
    ws->v0c[p]   = v0c;
    ws->valid[p] = valid ? 1 : 0;
    if (valid) atomicMin(&ws->first, p);
}

// ---------------- Kernel 3: gather stored row + its norm^2 ----------------
__global__ void k_stored(const float* __restrict__ feats, Ws* __restrict__ ws) {
    __shared__ float red[256];
    int c = threadIdx.x;
    int f = ws->first;
    if (f == 0x7FFFFFFF) f = 0;
    int v  = f % N_VIEWS;
    int u0 = ws->u0c[f];
    int v0 = ws->v0c[f];
    const float* b = feats + (size_t)v * CHW + (size_t)c * HW + (size_t)v0 * WF + u0;
    float val = 0.25f * (b[0] + b[1] + b[WF] + b[WF + 1]);
    ws->stored[c] = val;
    red[c] = val * val;
    __syncthreads();
#pragma unroll
    for (int s = 128; s > 0; s >>= 1) {
        if (c < s) red[c] += red[c + s];
        __syncthreads();
    }
    if (c == 0) ws->stored_norm2 = red[0];
}

// ---------------- Kernel 4: fused gather + WMMA f32 dot products ----------------
// 24 blocks x 128 threads (4 waves); each wave owns 16 rows, K tiled by 4 via
// v_wmma_f32_16x16x4_f32.  B = stored chunk broadcast across all 16 columns,
// so D[m][n] == dot(row m, stored) for every n.
__global__ void __launch_bounds__(128)
k_dots(const float* __restrict__ feats, Ws* __restrict__ ws) {
    __shared__ float ls[C_FEAT];
    for (int i = threadIdx.x; i < C_FEAT; i += 128) ls[i] = ws->stored[i];
    __syncthreads();

    const int wave    = threadIdx.x >> 5;
    const int lane    = threadIdx.x & 31;
    const int rowbase = (blockIdx.x * 4 + wave) * 16;
    const int row     = rowbase + (lane & 15);
    const int khalf   = lane >> 4;            // 0: K=k0,k0+1   1: K=k0+2,k0+3

    const int v  = row % N_VIEWS;
    const int u0 = ws->u0c[row];
    const int v0 = ws->v0c[row];
    const float* fb = feats + (size_t)v * CHW + (size_t)v0 * WF + u0;

    v8f  acc = {};
    float n2 = 0.f;

    for (int ch = 0; ch < C_FEAT / 4; ++ch) {
        const int k = ch * 4 + khalf * 2;
        const float* p0 = fb + (size_t)k * HW;
        const float* p1 = fb + (size_t)(k + 1) * HW;
        v2f a;
        a.x = 0.25f * (p0[0] + p0[1] + p0[WF] + p0[WF + 1]);
        a.y = 0.25f * (p1[0] + p1[1] + p1[WF] + p1[WF + 1]);
        v2f b;
        b.x = ls[k];
        b.y = ls[k + 1];
        // D = A(16x4) * B(4x16) + C ; fp32 WMMA
        acc = __builtin_amdgcn_wmma_f32_16x16x4_f32(
            /*neg_a=*/false, a, /*neg_b=*/false, b,
            /*c_mod=*/(short)0, acc, /*reuse_a=*/false, /*reuse_b=*/false);
        n2 += a.x * a.x + a.y * a.y;
    }

    // fold the two K-halves of the row norm (lanes l and l+16 hold complements)
    n2 += __shfl_xor(n2, 16, 32);
    if (lane < 16) ws->norm2[row] = n2;

    // D layout: VGPR r, lane 0 -> D[r][0] (rows 0..7); lane 16 -> D[8+r][0]
    if (lane == 0) {
#pragma unroll
        for (int r = 0; r < 8; ++r) ws->dots[rowbase + r] = acc[r];
    }
    if (lane == 16) {
#pragma unroll
        for (int r = 0; r < 8; ++r) ws->dots[rowbase + 8 + r] = acc[r];
    }
}

// ---------------- Kernel 5: finalize masked mean ----------------
__global__ void k_final(const Ws* __restrict__ ws, float* __restrict__ out) {
    __shared__ float ssum[256];
    __shared__ int   scnt[256];
    int t = threadIdx.x;
    int f = ws->first;
    if (f == 0x7FFFFFFF) f = 0;
    float sn = fmaxf(sqrtf(ws->stored_norm2), EPSV);

    float lsum = 0.f;
    int   lcnt = 0;
    for (int p = t; p < P_TOT; p += 256) {
        if (ws->valid[p] && p != f) {
            float nrm = fmaxf(sqrtf(ws->norm2[p]), EPSV);
            float cs  = ws->dots[p] / (nrm * sn);
            lsum += 1.f - cs;
            lcnt += 1;
        }
    }
    ssum[t] = lsum;
    scnt[t] = lcnt;
    __syncthreads();
#pragma unroll
    for (int s = 128; s > 0; s >>= 1) {
        if (t < s) { ssum[t] += ssum[t + s]; scnt[t] += scnt[t + s]; }
        __syncthreads();
    }
    if (t == 0) {
        int   cnt = scnt[0];
        float tot = ssum[0];
        out[0] = (cnt > 0) ? tot / (float)(cnt > 1 ? cnt : 1) : 0.f;
    }
}

extern "C" void kernel_launch(void* const* d_in, const int* in_sizes, int n_in,
                              void* d_out, int out_size, void* d_ws, size_t ws_size,
                              hipStream_t stream) {
    const float* boxes = (const float*)d_in[0];
    const float* feats = (const float*)d_in[1];
    const float* Km    = (const float*)d_in[2];
    const float* Tm    = (const float*)d_in[3];
    const int*   img   = (const int*)d_in[4];
    Ws*    ws  = (Ws*)d_ws;
    float* out = (float*)d_out;

    k_init   <<<1, 32, 0, stream>>>(ws);
    k_project<<<(P_TOT + 255) / 256, 256, 0, stream>>>(boxes, Km, Tm, img, ws);
    k_stored <<<1, 256, 0, stream>>>(feats, ws);
    k_dots   <<<P_TOT / (4 * 16), 128, 0, stream>>>(feats, ws);
    k_final  <<<1, 256, 0, stream>>>(ws, out);
}